// UnidirectionalMambaBlock_4544075399323
// MI455X (gfx1250) — compile-verified
//
#include <hip/hip_runtime.h>

#define B_SZ 4
#define L_SZ 2048
#define DM   512
#define DI   1024
#define DS   16
#define DTR  32
#define NTOK (B_SZ * L_SZ)   // 8192 tokens

typedef float v2f __attribute__((ext_vector_type(2)));
typedef float v8f __attribute__((ext_vector_type(8)));

// ---------------------------------------------------------------- utilities
__device__ __forceinline__ float wave_sum(float v) {
#pragma unroll
  for (int off = 16; off > 0; off >>= 1)
    v += __shfl_xor(v, off, 32);
  return v;
}

__device__ __forceinline__ float sigmoidf_fast(float x) {
  return 1.f / (1.f + __expf(-x));
}

// ---------------------------------------------------------------- LayerNorm
// One wave per token (512 features -> 16 per lane). res==nullptr for LN1,
// res=h2 (out-proj output) added to x before LN2.
__global__ __launch_bounds__(256) void ln512_kernel(
    const float* __restrict__ x, const float* __restrict__ res,
    const float* __restrict__ w, const float* __restrict__ bias,
    float* __restrict__ out)
{
  int wave = threadIdx.x >> 5;
  int lane = threadIdx.x & 31;
  int tok  = blockIdx.x * 8 + wave;
  const float* xin = x + (size_t)tok * DM;
  const float* rin = res ? res + (size_t)tok * DM : nullptr;

  float v[16];
  float s = 0.f;
#pragma unroll
  for (int i = 0; i < 16; ++i) {
    float t = xin[i * 32 + lane];
    if (rin) t += rin[i * 32 + lane];
    v[i] = t;
    s += t;
  }
  s = wave_sum(s);
  float mean = s * (1.f / (float)DM);
  float q = 0.f;
#pragma unroll
  for (int i = 0; i < 16; ++i) { float d = v[i] - mean; q += d * d; }
  q = wave_sum(q);
  float rstd = rsqrtf(q * (1.f / (float)DM) + 1e-5f);

  float* o = out + (size_t)tok * DM;
#pragma unroll
  for (int i = 0; i < 16; ++i) {
    int c = i * 32 + lane;
    o[c] = (v[i] - mean) * rstd * w[c] + bias[c];
  }
}

// ---------------------------------------------------------------- fp32 WMMA GEMM
// C[M,N] = A[M,K] @ B[K,N], row-major; all tile dims divide exactly.
// Block: 256 threads = 8 waves; block tile 64(M) x TN(N); K staged 32-deep.
// B is stored TRANSPOSED in LDS (sBt[col][k], stride 36) so both A and B
// fragments are single conflict-free ds_load_b64 of K-adjacent pairs.
// Wave w: M sub-tile (w&3)*16, N sub-tile (w>>2)*(TN/2) -> TN/32 accumulators.
// Next K-stage's global loads are register-prefetched under the WMMA loop.
#define GT_M 64
#define GT_K 32
#define LDSPAD 36   // conflict-free fragment reads, keeps 8/16B alignment

template<int TN>
__global__ __launch_bounds__(256) void gemm_f32_wmma(
    const float* __restrict__ A, const float* __restrict__ Bm,
    float* __restrict__ C, int K, int lda, int ldb, int ldc)
{
  constexpr int NACC = TN / 32;   // accumulators per wave
  constexpr int BF4  = TN / 32;   // float4 B loads per thread per stage
  __shared__ float sA [GT_M][LDSPAD];   // 9 KB
  __shared__ float sBt[TN]  [LDSPAD];   // 9/18 KB (transposed B)

  const int bm   = blockIdx.y * GT_M;
  const int bn   = blockIdx.x * TN;
  const int tid  = threadIdx.x;
  const int wave = tid >> 5;
  const int lane = tid & 31;
  const int wm   = (wave & 3) * 16;
  const int wn   = (wave >> 2) * (TN / 2);
  const int half = lane >> 4;   // 0: K+0/K+1, 1: K+2/K+3
  const int lr   = lane & 15;   // M index (A) / N index (B,C)

  // staging indices: A 64x32 (8 floats/thread), B 32xTN (TN/8 floats/thread)
  const int ar = tid >> 2, ac = (tid & 3) * 8;
  const int br = tid >> 3, bc = (tid & 7) * (TN / 8);

  const v8f vzero = {0.f,0.f,0.f,0.f,0.f,0.f,0.f,0.f};
  v8f acc[NACC];
#pragma unroll
  for (int j = 0; j < NACC; ++j) acc[j] = vzero;

  float4 apre[2];
  float4 bpre[BF4];
  {   // prefetch stage k0 = 0
    const float* as = A + (size_t)(bm + ar) * lda + ac;
    apre[0] = *(const float4*)as;
    apre[1] = *(const float4*)(as + 4);
    const float* bs = Bm + (size_t)br * ldb + bn + bc;
#pragma unroll
    for (int j = 0; j < BF4; ++j) bpre[j] = *(const float4*)(bs + 4 * j);
  }

  for (int k0 = 0; k0 < K; k0 += GT_K) {
    // commit prefetched stage to LDS (B goes in transposed)
    *(float4*)&sA[ar][ac]     = apre[0];
    *(float4*)&sA[ar][ac + 4] = apre[1];
#pragma unroll
    for (int j = 0; j < BF4; ++j) {
      float4 v = bpre[j];
      int c = bc + 4 * j;
      sBt[c + 0][br] = v.x;  sBt[c + 1][br] = v.y;
      sBt[c + 2][br] = v.z;  sBt[c + 3][br] = v.w;
    }
    __syncthreads();

    // register-prefetch next stage while WMMAs run
    if (k0 + GT_K < K) {
      const float* as = A + (size_t)(bm + ar) * lda + (k0 + GT_K + ac);
      apre[0] = *(const float4*)as;
      apre[1] = *(const float4*)(as + 4);
      const float* bs = Bm + (size_t)(k0 + GT_K + br) * ldb + bn + bc;
#pragma unroll
      for (int j = 0; j < BF4; ++j) bpre[j] = *(const float4*)(bs + 4 * j);
    }

#pragma unroll
    for (int kk = 0; kk < GT_K; kk += 4) {
      const int ka = kk + half * 2;
      v2f af = *(const v2f*)&sA[wm + lr][ka];
#pragma unroll
      for (int j = 0; j < NACC; ++j) {
        v2f bf = *(const v2f*)&sBt[wn + j * 16 + lr][ka];
        acc[j] = __builtin_amdgcn_wmma_f32_16x16x4_f32(false, af, false, bf,
                                                       (short)0, acc[j], false, false);
      }
    }
    __syncthreads();
  }

  // C/D layout: VGPR r -> rows (r, r+8); N = lane&15 within each 16-col tile.
#pragma unroll
  for (int j = 0; j < NACC; ++j) {
#pragma unroll
    for (int r = 0; r < 8; ++r) {
      int row = bm + wm + r + half * 8;
      C[(size_t)row * ldc + bn + wn + j * 16 + lr] = acc[j][r];
    }
  }
}

// ---------------------------------------------------------------- depthwise causal conv + SiLU
// uc[b,l,c] = silu( bias[c] + sum_k u[b,l-3+k,c] * w[c,k] ),  u = xz[..., :DI]
__global__ __launch_bounds__(256) void conv_silu_kernel(
    const float* __restrict__ xz, const float* __restrict__ cw,
    const float* __restrict__ cb, float* __restrict__ uc)
{
  int idx = blockIdx.x * 256 + threadIdx.x;   // over NTOK*DI, exact
  int c  = idx & (DI - 1);
  int bl = idx >> 10;
  int l  = bl & (L_SZ - 1);
  int b  = bl >> 11;

  float w0 = cw[c * 4 + 0], w1 = cw[c * 4 + 1], w2 = cw[c * 4 + 2], w3 = cw[c * 4 + 3];
  const float* base = xz + ((size_t)b * L_SZ) * (2 * DI) + c;
  float acc = cb[c];
  if (l >= 3) acc += base[(size_t)(l - 3) * (2 * DI)] * w0;
  if (l >= 2) acc += base[(size_t)(l - 2) * (2 * DI)] * w1;
  if (l >= 1) acc += base[(size_t)(l - 1) * (2 * DI)] * w2;
  acc += base[(size_t)l * (2 * DI)] * w3;
  uc[idx] = acc * sigmoidf_fast(acc);
}

// ---------------------------------------------------------------- selective scan (+softplus dt, +gating)
// One thread per (batch, channel). dt_lin buffer is overwritten in place with
// the gated output y (each element read then written by its owning thread).
__global__ __launch_bounds__(256) void scan_kernel(
    const float* __restrict__ xz,     // [NTOK, 2*DI]  (z at col DI+e)
    const float* __restrict__ uc,     // [NTOK, DI]
    const float* __restrict__ xdbl,   // [NTOK, 64]  (B at 32+n, C at 48+n)
    float* __restrict__ dt_y,         // in: dt_lin [NTOK, DI]; out: gated y
    const float* __restrict__ b_dt, const float* __restrict__ A_log,
    const float* __restrict__ D_param)
{
  int b = blockIdx.y;
  int e = blockIdx.x * 256 + threadIdx.x;

  float A[DS];
#pragma unroll
  for (int n = 0; n < DS; ++n) A[n] = -__expf(A_log[e * DS + n]);
  const float bd = b_dt[e];
  const float Dp = D_param[e];

  float h[DS];
#pragma unroll
  for (int n = 0; n < DS; ++n) h[n] = 0.f;

  const size_t tok0 = (size_t)b * L_SZ;
  for (int l = 0; l < L_SZ; ++l) {
    const size_t t = tok0 + l;
    float dtl = dt_y[t * DI + e] + bd;
    dtl = (dtl > 20.f) ? dtl : log1pf(__expf(dtl));   // softplus
    const float ut = uc[t * DI + e];
    const float du = dtl * ut;

    const float4* Bp = (const float4*)(xdbl + t * 64 + 32);
    const float4* Cp = (const float4*)(xdbl + t * 64 + 48);
    float ysum = 0.f;
#pragma unroll
    for (int q = 0; q < 4; ++q) {
      float4 Bv = Bp[q];
      float4 Cv = Cp[q];
      int n = q * 4;
      h[n + 0] = __expf(dtl * A[n + 0]) * h[n + 0] + du * Bv.x;  ysum += h[n + 0] * Cv.x;
      h[n + 1] = __expf(dtl * A[n + 1]) * h[n + 1] + du * Bv.y;  ysum += h[n + 1] * Cv.y;
      h[n + 2] = __expf(dtl * A[n + 2]) * h[n + 2] + du * Bv.z;  ysum += h[n + 2] * Cv.z;
      h[n + 3] = __expf(dtl * A[n + 3]) * h[n + 3] + du * Bv.w;  ysum += h[n + 3] * Cv.w;
    }
    const float z = xz[t * (2 * DI) + DI + e];
    dt_y[t * DI + e] = (ysum + ut * Dp) * (z * sigmoidf_fast(z));
  }
}

// ---------------------------------------------------------------- launch
extern "C" void kernel_launch(void* const* d_in, const int* in_sizes, int n_in,
                              void* d_out, int out_size, void* d_ws, size_t ws_size,
                              hipStream_t stream) {
  (void)in_sizes; (void)n_in; (void)out_size; (void)ws_size;

  const float* x      = (const float*)d_in[0];
  const float* ln1_w  = (const float*)d_in[1];
  const float* ln1_b  = (const float*)d_in[2];
  const float* ln2_w  = (const float*)d_in[3];
  const float* ln2_b  = (const float*)d_in[4];
  const float* W_in   = (const float*)d_in[5];
  const float* conv_w = (const float*)d_in[6];
  const float* conv_b = (const float*)d_in[7];
  const float* W_xprj = (const float*)d_in[8];
  const float* W_dt   = (const float*)d_in[9];
  const float* b_dt   = (const float*)d_in[10];
  const float* A_log  = (const float*)d_in[11];
  const float* D_par  = (const float*)d_in[12];
  const float* W_out  = (const float*)d_in[13];
  float* out = (float*)d_out;

  // workspace layout (floats); total 38,273,024 floats ~= 153 MB
  float* ws    = (float*)d_ws;
  float* h_ln  = ws;                       // 8192*512   = 4,194,304 (reused for h2)
  float* xz    = ws + 4194304;             // 8192*2048  = 16,777,216
  float* uc    = ws + 20971520;            // 8192*1024  = 8,388,608
  float* xdbl  = ws + 29360128;            // 8192*64    = 524,288
  float* dt_y  = ws + 29884416;            // 8192*1024  = 8,388,608 (dt_lin, then y)
  float* h2    = h_ln;                     // reuse: h_ln dead after GEMM1

  // 1) LN1
  ln512_kernel<<<NTOK / 8, 256, 0, stream>>>(x, nullptr, ln1_w, ln1_b, h_ln);

  // 2) xz = h_ln @ W_in   [8192x512]@[512x2048]
  gemm_f32_wmma<128><<<dim3(2 * DI / 128, NTOK / GT_M), 256, 0, stream>>>(
      h_ln, W_in, xz, DM, DM, 2 * DI, 2 * DI);

  // 3) depthwise conv + SiLU on u half
  conv_silu_kernel<<<(NTOK * DI) / 256, 256, 0, stream>>>(xz, conv_w, conv_b, uc);

  // 4) x_dbl = uc @ W_xproj   [8192x1024]@[1024x64]
  gemm_f32_wmma<64><<<dim3(1, NTOK / GT_M), 256, 0, stream>>>(
      uc, W_xprj, xdbl, DI, DI, 64, 64);

  // 5) dt_lin = x_dbl[:, :32] @ W_dt   [8192x32]@[32x1024]
  gemm_f32_wmma<128><<<dim3(DI / 128, NTOK / GT_M), 256, 0, stream>>>(
      xdbl, W_dt, dt_y, DTR, 64, DI, DI);

  // 6) selective scan + softplus + gating (writes gated y into dt_y in place)
  scan_kernel<<<dim3(DI / 256, B_SZ), 256, 0, stream>>>(
      xz, uc, xdbl, dt_y, b_dt, A_log, D_par);

  // 7) h2 = y @ W_out   [8192x1024]@[1024x512]
  gemm_f32_wmma<128><<<dim3(DM / 128, NTOK / GT_M), 256, 0, stream>>>(
      dt_y, W_out, h2, DI, DI, DM, DM);

  // 8) out = LN2(x + h2)
  ln512_kernel<<<NTOK / 8, 256, 0, stream>>>(x, h2, ln2_w, ln2_b, out);
}